// IntraAgg_5119601017180
// MI455X (gfx1250) — compile-verified
//
#include <hip/hip_runtime.h>
#include <hip/hip_bf16.h>

typedef __attribute__((ext_vector_type(2))) float v2f;
typedef __attribute__((ext_vector_type(8))) float v8f;

#define D_FEAT 96

// ---------------------------------------------------------------------------
// Kernel A: build CSR-style segment start offsets from sorted seg_ids.
// start[b] = first edge index with seg >= b ; start has B+1 entries.
// ---------------------------------------------------------------------------
__global__ void __launch_bounds__(256)
build_starts_kernel(const int* __restrict__ seg, int E, int B,
                    int* __restrict__ start) {
    int e = blockIdx.x * blockDim.x + threadIdx.x;
    if (e > E) return;
    if (e == 0) {
        int s0 = seg[0];
        for (int s = 0; s <= s0 && s <= B; ++s) start[s] = 0;
    } else if (e == E) {
        int sl = seg[E - 1];
        for (int s = sl + 1; s <= B; ++s) start[s] = E;
    } else {
        int sp = seg[e - 1];
        int sc = seg[e];
        for (int s = sp + 1; s <= sc && s <= B; ++s) start[s] = e;
    }
}

// ---------------------------------------------------------------------------
// Kernel B: one wave32 per 16-segment tile. Segment-sum via one-hot mask
// matmul on V_WMMA_F32_16X16X4_F32 (exact f32), then mean + concat epilogue.
// ---------------------------------------------------------------------------
__global__ void __launch_bounds__(256)
agg_wmma_kernel(const float* __restrict__ emb,
                const float* __restrict__ selff,
                const int*   __restrict__ nidx,
                const int*   __restrict__ seg,
                const int*   __restrict__ start,
                float* __restrict__ out,
                int B, int E) {
    const int lane = threadIdx.x & 31;
    const int wave = blockIdx.x * (blockDim.x >> 5) + (threadIdx.x >> 5);
    const int b0   = wave * 16;
    if (b0 >= B) return;                 // wave-uniform exit: EXEC stays full

    const int half = lane >> 4;          // 0 -> edges e+0,e+1 ; 1 -> e+2,e+3
    const int lid  = lane & 15;          // column index / segment-in-tile
    const int myseg = b0 + lid;          // segment this lane's A-row matches

    const int bHi  = (b0 + 16 <= B) ? (b0 + 16) : B;
    const int eBeg = start[b0];
    const int eEnd = start[bHi];

    v8f acc0 = {}, acc1 = {}, acc2 = {}, acc3 = {}, acc4 = {}, acc5 = {};

    int e = eBeg;

    // ---- fast path: wave-uniform bound, all 4 chunk edges are in-range ----
#pragma unroll 2
    for (; e < eEnd && (e + 4 <= E); e += 4) {
        const int k0 = e + half * 2;     // K=0 (lower half) / K=2 (upper half)
        const int k1 = k0 + 1;           // K=1 / K=3

        const float a0 = (seg[k0] == myseg) ? 1.0f : 0.0f;
        const float a1 = (seg[k1] == myseg) ? 1.0f : 0.0f;
        const unsigned o0 = (unsigned)nidx[k0] * 96u + (unsigned)lid;
        const unsigned o1 = (unsigned)nidx[k1] * 96u + (unsigned)lid;
        const float* r0 = emb + o0;
        const float* r1 = emb + o1;

        v2f a; a.x = a0; a.y = a1;       // A: VGPR0=K0/K2, VGPR1=K1/K3
        v2f b;

        b.x = r0[0];  b.y = r1[0];
        acc0 = __builtin_amdgcn_wmma_f32_16x16x4_f32(false, a, false, b,
                                                     (short)0, acc0, false, false);
        b.x = r0[16]; b.y = r1[16];
        acc1 = __builtin_amdgcn_wmma_f32_16x16x4_f32(false, a, false, b,
                                                     (short)0, acc1, false, false);
        b.x = r0[32]; b.y = r1[32];
        acc2 = __builtin_amdgcn_wmma_f32_16x16x4_f32(false, a, false, b,
                                                     (short)0, acc2, false, false);
        b.x = r0[48]; b.y = r1[48];
        acc3 = __builtin_amdgcn_wmma_f32_16x16x4_f32(false, a, false, b,
                                                     (short)0, acc3, false, false);
        b.x = r0[64]; b.y = r1[64];
        acc4 = __builtin_amdgcn_wmma_f32_16x16x4_f32(false, a, false, b,
                                                     (short)0, acc4, false, false);
        b.x = r0[80]; b.y = r1[80];
        acc5 = __builtin_amdgcn_wmma_f32_16x16x4_f32(false, a, false, b,
                                                     (short)0, acc5, false, false);
    }

    // ---- tail: at most one chunk, may touch edges >= E (last tile only) ---
    for (; e < eEnd; e += 4) {
        const int k0 = e + half * 2;
        const int k1 = k0 + 1;
        // clamp per-lane, zero the mask for out-of-range edges; a=0 kills
        // any contribution from the clamped (edge E-1) embedding load.
        const int c0 = (k0 < E) ? k0 : (E - 1);
        const int c1 = (k1 < E) ? k1 : (E - 1);
        const float a0 = (k0 < E && seg[c0] == myseg) ? 1.0f : 0.0f;
        const float a1 = (k1 < E && seg[c1] == myseg) ? 1.0f : 0.0f;
        const unsigned o0 = (unsigned)nidx[c0] * 96u + (unsigned)lid;
        const unsigned o1 = (unsigned)nidx[c1] * 96u + (unsigned)lid;
        const float* r0 = emb + o0;
        const float* r1 = emb + o1;

        v2f a; a.x = a0; a.y = a1;
        v2f b;

        b.x = r0[0];  b.y = r1[0];
        acc0 = __builtin_amdgcn_wmma_f32_16x16x4_f32(false, a, false, b,
                                                     (short)0, acc0, false, false);
        b.x = r0[16]; b.y = r1[16];
        acc1 = __builtin_amdgcn_wmma_f32_16x16x4_f32(false, a, false, b,
                                                     (short)0, acc1, false, false);
        b.x = r0[32]; b.y = r1[32];
        acc2 = __builtin_amdgcn_wmma_f32_16x16x4_f32(false, a, false, b,
                                                     (short)0, acc2, false, false);
        b.x = r0[48]; b.y = r1[48];
        acc3 = __builtin_amdgcn_wmma_f32_16x16x4_f32(false, a, false, b,
                                                     (short)0, acc3, false, false);
        b.x = r0[64]; b.y = r1[64];
        acc4 = __builtin_amdgcn_wmma_f32_16x16x4_f32(false, a, false, b,
                                                     (short)0, acc4, false, false);
        b.x = r0[80]; b.y = r1[80];
        acc5 = __builtin_amdgcn_wmma_f32_16x16x4_f32(false, a, false, b,
                                                     (short)0, acc5, false, false);
    }

    // Epilogue: C element r maps to row b0 + r + 8*half, col = lid (+16*tile).
    float inv[8];
#pragma unroll
    for (int r = 0; r < 8; ++r) {
        int b = b0 + half * 8 + r;
        int cnt = 1;
        if (b < B) {
            cnt = start[b + 1] - start[b];
            if (cnt < 1) cnt = 1;
        }
        inv[r] = 1.0f / (float)cnt;
    }

#pragma unroll
    for (int r = 0; r < 8; ++r) {
        int b = b0 + half * 8 + r;
        if (b >= B) continue;            // after all WMMAs: divergence OK
        const size_t obase = (size_t)b * (2 * D_FEAT);
        const size_t sbase = (size_t)b * D_FEAT;
        float f;
        f = acc0[r] * inv[r];
        out[obase +  0 + lid] = f;
        out[obase + D_FEAT +  0 + lid] = selff[sbase +  0 + lid] - f;
        f = acc1[r] * inv[r];
        out[obase + 16 + lid] = f;
        out[obase + D_FEAT + 16 + lid] = selff[sbase + 16 + lid] - f;
        f = acc2[r] * inv[r];
        out[obase + 32 + lid] = f;
        out[obase + D_FEAT + 32 + lid] = selff[sbase + 32 + lid] - f;
        f = acc3[r] * inv[r];
        out[obase + 48 + lid] = f;
        out[obase + D_FEAT + 48 + lid] = selff[sbase + 48 + lid] - f;
        f = acc4[r] * inv[r];
        out[obase + 64 + lid] = f;
        out[obase + D_FEAT + 64 + lid] = selff[sbase + 64 + lid] - f;
        f = acc5[r] * inv[r];
        out[obase + 80 + lid] = f;
        out[obase + D_FEAT + 80 + lid] = selff[sbase + 80 + lid] - f;
    }
}

// ---------------------------------------------------------------------------
extern "C" void kernel_launch(void* const* d_in, const int* in_sizes, int n_in,
                              void* d_out, int out_size, void* d_ws, size_t ws_size,
                              hipStream_t stream) {
    const float* emb   = (const float*)d_in[0];   // [N, 96]
    const float* selff = (const float*)d_in[1];   // [B, 96]
    const int*   nidx  = (const int*)d_in[2];     // [E]
    const int*   seg   = (const int*)d_in[3];     // [E] sorted
    float*       out   = (float*)d_out;           // [B, 192]

    const int E = in_sizes[2];
    const int B = in_sizes[1] / D_FEAT;

    int* start = (int*)d_ws;                      // B+1 ints, fully rewritten

    {
        int threads = 256;
        int blocks  = (E + 1 + threads - 1) / threads;
        build_starts_kernel<<<blocks, threads, 0, stream>>>(seg, E, B, start);
    }
    {
        int tiles         = (B + 15) / 16;        // one wave32 per tile
        int wavesPerBlock = 8;                    // 256 threads
        int blocks        = (tiles + wavesPerBlock - 1) / wavesPerBlock;
        agg_wmma_kernel<<<blocks, 256, 0, stream>>>(emb, selff, nidx, seg,
                                                    start, out, B, E);
    }
}